// MultiHeadComplexEMA_45638322487507
// MI455X (gfx1250) — compile-verified
//
#include <hip/hip_runtime.h>
#include <math.h>
#include <stdint.h>

typedef __attribute__((ext_vector_type(2))) float v2f;
typedef __attribute__((ext_vector_type(4))) float v4f;
typedef __attribute__((ext_vector_type(8))) float v8f;

#define B_   8
#define D_   1024
#define L_   4096
#define N_   16
#define T_   64          // chunk length
#define C_   64          // chunks per sequence (L_/T_)
#define COLS 512         // B_*C_ columns per channel

// workspace layout (floats)
static constexpr size_t K0 = 0;                          // k[d][64]
static constexpr size_t A0 = K0 + (size_t)D_ * 64;       // Amat[d][32][64]
static constexpr size_t G0 = A0 + (size_t)D_ * 32 * 64;  // Gmat[d][64][32]
static constexpr size_t Q0 = G0 + (size_t)D_ * 64 * 32;  // qT[d][32]
static constexpr size_t S0 = Q0 + (size_t)D_ * 32;       // S/H [d][col=512][n2=32]

// --- CDNA5 async direct-to-LDS copy (VGLOBAL async, tracked by ASYNCcnt) ----
__device__ __forceinline__ void async_ld_b128(void* lds_ptr, const float* gptr) {
  asm volatile("global_load_async_to_lds_b128 %0, %1, off"
               :: "v"((uint32_t)(uintptr_t)lds_ptr), "v"(gptr)
               : "memory");
}
__device__ __forceinline__ void wait_async0() {
  asm volatile("s_wait_asynccnt 0x0" ::: "memory");
}

// ---------------------------------------------------------------------------
// Kernel A: per-channel coefficients (tiny).
// ---------------------------------------------------------------------------
__global__ __launch_bounds__(64) void coeff_kernel(
    const float* __restrict__ alpha, const float* __restrict__ delta,
    const float* __restrict__ theta, const float* __restrict__ gamma,
    float* __restrict__ ws) {
  const int d = blockIdx.x;
  const int tid = threadIdx.x;
  __shared__ float qs[N_][T_ + 1][2];     // q^s, s=0..64
  __shared__ float pm[N_], gre[N_], gim[N_];

  if (tid < N_) {
    const int n = tid;
    const float a  = 1.f / (1.f + expf(-alpha[d * N_ + n]));
    const float dl = 1.f / (1.f + expf(-delta[d * N_ + n]));
    const float thb = (1.f / (1.f + expf(-theta[d]))) * (6.283185307179586f / (float)N_);
    const float th = (float)(n + 1) * thb;
    const float r  = 1.f - a * dl;        // in (0,1)
    const float lr = logf(r);
    pm[n]  = a;
    gre[n] = gamma[(d * N_ + n) * 2 + 0] * 0.25f;
    gim[n] = gamma[(d * N_ + n) * 2 + 1] * 0.25f;
    for (int s = 0; s <= T_; ++s) {
      const float mag = expf(lr * (float)s);
      const float ang = th * (float)s;
      qs[n][s][0] = mag * cosf(ang);
      qs[n][s][1] = mag * sinf(ang);
    }
  }
  __syncthreads();

  {  // k[s] = Re(sum_n p g q^s)
    float acc = 0.f;
    for (int n = 0; n < N_; ++n)
      acc += pm[n] * (gre[n] * qs[n][tid][0] - gim[n] * qs[n][tid][1]);
    ws[K0 + (size_t)d * 64 + tid] = acc;
  }
  {  // Amat[2n+ri][t] = p * {Re,Im}(q^{63-t})
    float* Am = ws + A0 + (size_t)d * 2048;
    for (int i = 0; i < 32; ++i) {
      const int e  = tid + i * 64;
      const int n2 = e >> 6, t = e & 63;
      const int n  = n2 >> 1;
      Am[e] = pm[n] * qs[n][63 - t][n2 & 1];
    }
  }
  {  // Gmat[t][2n] = Re(g q^{t+1}); Gmat[t][2n+1] = -Im(g q^{t+1})
    float* Gm = ws + G0 + (size_t)d * 2048;
    for (int i = 0; i < 32; ++i) {
      const int e  = tid + i * 64;
      const int t  = e >> 5, n2 = e & 31;
      const int n  = n2 >> 1;
      const float qr = qs[n][t + 1][0], qi = qs[n][t + 1][1];
      Gm[e] = (n2 & 1) ? -(gre[n] * qi + gim[n] * qr)
                       :  (gre[n] * qr - gim[n] * qi);
    }
  }
  if (tid < 32) {  // q^64
    const int n = tid >> 1;
    ws[Q0 + (size_t)d * 32 + tid] = qs[n][T_][tid & 1];
  }
}

// ---------------------------------------------------------------------------
// Kernel B: S[d][col][n2] = Amat(32x64) @ X(64x512) via V_WMMA_F32_16X16X4_F32
// One workgroup = (d, 128-col block); 8 waves, one 16-col tile each, 2 row tiles.
// ---------------------------------------------------------------------------
__global__ __launch_bounds__(256) void collect_kernel(
    const float* __restrict__ x, float* __restrict__ ws) {
  const int d  = blockIdx.x;
  const int cb = blockIdx.y;            // 0..3
  const int tid = threadIdx.x;
  __shared__ float lx[128 * 68];        // X staged col-major, pad 4
  __shared__ float lam[32 * 68];        // Amat staged, pad 4

  // stage X (8192 floats) and Amat (2048 floats) via async direct-to-LDS b128
  for (int i = 0; i < 8; ++i) {
    const int f   = tid + i * 256;
    const int col = f >> 4;             // 0..127
    const int t4  = (f & 15) << 2;
    const int j   = cb * 128 + col;
    const int b   = j >> 6, c = j & 63;
    async_ld_b128(&lx[col * 68 + t4],
                  x + (((size_t)b * D_ + d) * L_ + c * T_ + t4));
  }
  {
    const float* Am = ws + A0 + (size_t)d * 2048;
    for (int i = 0; i < 2; ++i) {
      const int f  = tid + i * 256;
      const int n2 = f >> 4;
      const int t4 = (f & 15) << 2;
      async_ld_b128(&lam[n2 * 68 + t4], Am + n2 * 64 + t4);
    }
  }
  wait_async0();
  __syncthreads();

  const int wave = tid >> 5, lane = tid & 31;
  const int m    = lane & 15;           // A-row / B-col / C-col index
  const int khi  = (lane >> 4) << 1;    // K sub-offset for this half-wave
  const int col  = wave * 16 + m;

  v8f acc0 = {}; v8f acc1 = {};
  for (int kk = 0; kk < 16; ++kk) {
    const int s0 = kk * 4 + khi;
    v2f bf; bf.x = lx[col * 68 + s0]; bf.y = lx[col * 68 + s0 + 1];
    v2f a0; a0.x = lam[m * 68 + s0];        a0.y = lam[m * 68 + s0 + 1];
    v2f a1; a1.x = lam[(16 + m) * 68 + s0]; a1.y = lam[(16 + m) * 68 + s0 + 1];
    acc0 = __builtin_amdgcn_wmma_f32_16x16x4_f32(false, a0, false, bf, (short)0, acc0, false, false);
    acc1 = __builtin_amdgcn_wmma_f32_16x16x4_f32(false, a1, false, bf, (short)0, acc1, false, false);
  }

  // store: S[d][col][n2]; each lane owns 8 consecutive n2 per accumulator
  float* Sp = ws + S0 + ((size_t)d * COLS + (size_t)cb * 128) * 32;
  const int rb = (lane >> 4) * 8;
  v4f w;
  w.x = acc0[0]; w.y = acc0[1]; w.z = acc0[2]; w.w = acc0[3];
  *(v4f*)(Sp + (size_t)col * 32 + rb) = w;
  w.x = acc0[4]; w.y = acc0[5]; w.z = acc0[6]; w.w = acc0[7];
  *(v4f*)(Sp + (size_t)col * 32 + rb + 4) = w;
  w.x = acc1[0]; w.y = acc1[1]; w.z = acc1[2]; w.w = acc1[3];
  *(v4f*)(Sp + (size_t)col * 32 + 16 + rb) = w;
  w.x = acc1[4]; w.y = acc1[5]; w.z = acc1[6]; w.w = acc1[7];
  *(v4f*)(Sp + (size_t)col * 32 + 16 + rb + 4) = w;
}

// ---------------------------------------------------------------------------
// Kernel C: per-(b,d,n) scan over 64 chunks: h <- q^64 h + S_c.
// Overwrites S slot c with the chunk-entry state H_c (in place); lanes 0-15
// cover one contiguous 128B line per step (coalesced float2 accesses).
// ---------------------------------------------------------------------------
__global__ __launch_bounds__(256) void scan_kernel(
    const float* __restrict__ hx, float* __restrict__ ws,
    float* __restrict__ hout, int write_h) {
  const int t = blockIdx.x * blockDim.x + threadIdx.x;  // 0..131071
  const int n = t & 15;
  const int d = (t >> 4) & (D_ - 1);
  const int b = t >> 14;
  float hr = hx[(((size_t)b * D_ + d) * N_ + n) * 2 + 0];
  float hi = hx[(((size_t)b * D_ + d) * N_ + n) * 2 + 1];
  const float qr = ws[Q0 + (size_t)d * 32 + n * 2 + 0];
  const float qi = ws[Q0 + (size_t)d * 32 + n * 2 + 1];
  float* Sp = ws + S0 + ((size_t)d * COLS + (size_t)b * C_) * 32 + 2 * n;
  for (int c = 0; c < C_; ++c) {
    v2f s = *(v2f*)(Sp + (size_t)c * 32);
    v2f h; h.x = hr; h.y = hi;
    *(v2f*)(Sp + (size_t)c * 32) = h;       // H_c = state entering chunk c
    const float nr = qr * hr - qi * hi + s.x;
    const float ni = qr * hi + qi * hr + s.y;
    hr = nr; hi = ni;
  }
  if (write_h) {
    hout[(((size_t)b * D_ + d) * N_ + n) * 2 + 0] = hr;
    hout[(((size_t)b * D_ + d) * N_ + n) * 2 + 1] = hi;
  }
}

// ---------------------------------------------------------------------------
// Kernel D: out = K_lowerToeplitz(64x64) @ X + Gmat(64x32) @ H + omega * x
// Both GEMMs accumulate into the same WMMA tiles.
// ---------------------------------------------------------------------------
__global__ __launch_bounds__(256) void output_kernel(
    const float* __restrict__ x, const float* __restrict__ omega,
    const float* __restrict__ ws, float* __restrict__ out) {
  const int d  = blockIdx.x;
  const int cb = blockIdx.y;
  const int tid = threadIdx.x;
  __shared__ float lx[128 * 68];    // X (64 rows x 128 cols), pad 4
  __shared__ float lh[128 * 36];    // H [col][n2], pad 4
  __shared__ float lgm[64 * 36];    // Gmat (64 x 32), pad 4
  __shared__ float lk[64];          // conv kernel taps

  for (int i = 0; i < 8; ++i) {
    const int f   = tid + i * 256;
    const int col = f >> 4;
    const int t4  = (f & 15) << 2;
    const int j   = cb * 128 + col;
    const int b   = j >> 6, c = j & 63;
    async_ld_b128(&lx[col * 68 + t4],
                  x + (((size_t)b * D_ + d) * L_ + c * T_ + t4));
  }
  {
    const float* Hp = ws + S0 + ((size_t)d * COLS + (size_t)cb * 128) * 32;
    for (int i = 0; i < 4; ++i) {
      const int f   = tid + i * 256;       // 0..1023 float4
      const int col = f >> 3;              // 8 float4 per col
      const int n4  = (f & 7) << 2;
      async_ld_b128(&lh[col * 36 + n4], Hp + (size_t)col * 32 + n4);
    }
  }
  {
    const float* Gm = ws + G0 + (size_t)d * 2048;
    for (int i = 0; i < 2; ++i) {
      const int f  = tid + i * 256;
      const int tt = f >> 3;
      const int n4 = (f & 7) << 2;
      async_ld_b128(&lgm[tt * 36 + n4], Gm + tt * 32 + n4);
    }
  }
  if (tid < 16) {
    async_ld_b128(&lk[tid * 4], ws + K0 + (size_t)d * 64 + tid * 4);
  }
  wait_async0();
  __syncthreads();

  const int wave = tid >> 5, lane = tid & 31;
  const int m    = lane & 15;
  const int khi  = (lane >> 4) << 1;
  const int col  = wave * 16 + m;

  v8f acc[4] = {v8f{}, v8f{}, v8f{}, v8f{}};

  // GEMM1: causal Toeplitz conv, K over 64
  for (int kk = 0; kk < 16; ++kk) {
    const int s0 = kk * 4 + khi;
    v2f bf; bf.x = lx[col * 68 + s0]; bf.y = lx[col * 68 + s0 + 1];
#pragma unroll
    for (int r = 0; r < 4; ++r) {
      const int tt = r * 16 + m;
      v2f af;
      af.x = (tt >= s0)     ? lk[tt - s0]     : 0.f;
      af.y = (tt >= s0 + 1) ? lk[tt - s0 - 1] : 0.f;
      acc[r] = __builtin_amdgcn_wmma_f32_16x16x4_f32(false, af, false, bf, (short)0, acc[r], false, false);
    }
  }
  // GEMM2: bias from chunk-entry states, K over 32 (n2)
  for (int kk = 0; kk < 8; ++kk) {
    const int s0 = kk * 4 + khi;
    v2f bf; bf.x = lh[col * 36 + s0]; bf.y = lh[col * 36 + s0 + 1];
#pragma unroll
    for (int r = 0; r < 4; ++r) {
      const int tt = r * 16 + m;
      v2f af; af.x = lgm[tt * 36 + s0]; af.y = lgm[tt * 36 + s0 + 1];
      acc[r] = __builtin_amdgcn_wmma_f32_16x16x4_f32(false, af, false, bf, (short)0, acc[r], false, false);
    }
  }

  // epilogue: + omega * x, contiguous 16B runs along t
  const float om = omega[d];
  const int j = cb * 128 + col;
  const int b = j >> 6, c = j & 63;
  float* op = out + ((size_t)b * D_ + d) * L_ + c * T_;
  const int rb = (lane >> 4) * 8;
#pragma unroll
  for (int r = 0; r < 4; ++r) {
    const int t0 = r * 16 + rb;
    v4f w0, w1;
#pragma unroll
    for (int v = 0; v < 4; ++v) {
      w0[v] = acc[r][v]     + om * lx[col * 68 + t0 + v];
      w1[v] = acc[r][v + 4] + om * lx[col * 68 + t0 + 4 + v];
    }
    *(v4f*)(op + t0)     = w0;
    *(v4f*)(op + t0 + 4) = w1;
  }
}

// ---------------------------------------------------------------------------
extern "C" void kernel_launch(void* const* d_in, const int* in_sizes, int n_in,
                              void* d_out, int out_size, void* d_ws, size_t ws_size,
                              hipStream_t stream) {
  const float* x     = (const float*)d_in[0];
  const float* hx    = (const float*)d_in[1];
  const float* alpha = (const float*)d_in[3];
  const float* delta = (const float*)d_in[4];
  const float* theta = (const float*)d_in[5];
  const float* gamma = (const float*)d_in[6];
  const float* omega = (const float*)d_in[7];
  float* out = (float*)d_out;
  float* ws  = (float*)d_ws;
  const int write_h = (out_size > (int)((size_t)B_ * D_ * L_)) ? 1 : 0;

  coeff_kernel<<<dim3(D_), dim3(64), 0, stream>>>(alpha, delta, theta, gamma, ws);
  collect_kernel<<<dim3(D_, 4), dim3(256), 0, stream>>>(x, ws);
  scan_kernel<<<dim3(512), dim3(256), 0, stream>>>(hx, ws, out + (size_t)B_ * D_ * L_, write_h);
  output_kernel<<<dim3(D_, 4), dim3(256), 0, stream>>>(x, omega, ws, out);
}